// NodeContrastiveLoss_5111011083049
// MI455X (gfx1250) — compile-verified
//
#include <hip/hip_runtime.h>
#include <hip/hip_bf16.h>
#include <math.h>

// Problem constants (from reference)
#define NB 2048
#define NA 256
#define NF 64
#define ND 128
#define TEMP_INV 10.0f   // 1 / 0.1
#define EPS 1e-8f

typedef __attribute__((ext_vector_type(2))) float v2f;
typedef __attribute__((ext_vector_type(8))) float v8f;

// LDS layout (floats), padded to avoid bank conflicts
#define MN_STRIDE 132                 // 128 + 4 -> 4-bank skew per row
#define MN_OFF    0
#define FN_OFF    (NF * MN_STRIDE)    // 8448
#define SIMS_STRIDE 68                // 64 + 4
#define SIMS_OFF  (FN_OFF + NF * MN_STRIDE)          // 16896
#define CNT_OFF   (SIMS_OFF + NF * SIMS_STRIDE)      // 21248
#define ACC_OFF   (CNT_OFF + NF)                     // 21312
#define SMEM_FLOATS (ACC_OFF + 2)                    // 21314 (~83.3 KB)

__global__ void ncl_init_ws(float* ws) {
    if (threadIdx.x < 2) ws[threadIdx.x] = 0.0f;
}

__global__ __launch_bounds__(256) void ncl_main(const float* __restrict__ atom_embed,
                                                const float* __restrict__ frag_embed,
                                                const int*   __restrict__ atom2frag,
                                                float* __restrict__ ws) {
    __shared__ __align__(16) float smem[SMEM_FLOATS];
    float* sm_mn   = smem + MN_OFF;     // [64][132] pooled sums -> normalized means
    float* sm_fn   = smem + FN_OFF;     // [64][132] fragment embeds -> normalized
    float* sm_sims = smem + SIMS_OFF;   // [64][68]
    float* sm_cnt  = smem + CNT_OFF;    // [64]
    float* sm_acc  = smem + ACC_OFF;    // [2] : {sum(per_frag), n_valid}

    const int tid  = threadIdx.x;
    const int lane = tid & 31;
    const int wave = tid >> 5;
    const int b    = blockIdx.x;

    // ---- Phase 0: zero sums/counts/acc, load fragment embeds into LDS ----
    for (int i = tid; i < NF * MN_STRIDE; i += 256) sm_mn[i] = 0.0f;
    if (tid < NF) sm_cnt[tid] = 0.0f;
    if (tid < 2)  sm_acc[tid] = 0.0f;

    const float* fsrc = frag_embed + (size_t)b * NF * ND;
    for (int idx = tid; idx < NF * (ND / 4); idx += 256) {
        int r  = idx >> 5;           // row 0..63
        int c4 = idx & 31;           // float4 index 0..31
        const float4 v = ((const float4*)(fsrc + (size_t)r * ND))[c4];
        *(float4*)(&sm_fn[r * MN_STRIDE + c4 * 4]) = v;
    }
    __syncthreads();

    // ---- Phase 1: scatter-add atoms into fragment sums (wave per atom) ----
    {
        const int* a2f = atom2frag + (size_t)b * NA;
        const float* asrc = atom_embed + (size_t)b * NA * ND;
        for (int i = 0; i < NA / 8; ++i) {           // 32 atoms per wave
            int a = wave * (NA / 8) + i;
            int frag = a2f[a];
            float* dst = &sm_mn[frag * MN_STRIDE];
            const float* src = asrc + (size_t)a * ND;
#pragma unroll
            for (int j = 0; j < 4; ++j) {
                int c = lane + 32 * j;               // conflict-free banks
                atomicAdd(&dst[c], src[c]);          // ds_add_f32
            }
            if (lane == 0) atomicAdd(&sm_cnt[frag], 1.0f);
        }
    }
    __syncthreads();

    // ---- Phase 2: normalize rows (wave per row): mn = mean/||mean||, fn = fn/||fn|| ----
    for (int r = wave; r < NF; r += 8) {
        // pooled means
        {
            float invc = 1.0f / fmaxf(sm_cnt[r], 1.0f);
            float* row = &sm_mn[r * MN_STRIDE];
            float v[4]; float ss = 0.0f;
#pragma unroll
            for (int j = 0; j < 4; ++j) { v[j] = row[lane + 32 * j] * invc; ss += v[j] * v[j]; }
#pragma unroll
            for (int m = 16; m >= 1; m >>= 1) ss += __shfl_xor(ss, m, 32);
            float inv = 1.0f / fmaxf(sqrtf(ss), EPS);
#pragma unroll
            for (int j = 0; j < 4; ++j) row[lane + 32 * j] = v[j] * inv;
        }
        // fragment embeds
        {
            float* row = &sm_fn[r * MN_STRIDE];
            float v[4]; float ss = 0.0f;
#pragma unroll
            for (int j = 0; j < 4; ++j) { v[j] = row[lane + 32 * j]; ss += v[j] * v[j]; }
#pragma unroll
            for (int m = 16; m >= 1; m >>= 1) ss += __shfl_xor(ss, m, 32);
            float inv = 1.0f / fmaxf(sqrtf(ss), EPS);
#pragma unroll
            for (int j = 0; j < 4; ++j) row[lane + 32 * j] = v[j] * inv;
        }
    }
    __syncthreads();

    // ---- Phase 3: sims[64][64] = mn @ fn^T via V_WMMA_F32_16X16X4_F32 ----
    // A (16x4 f32): lanes 0-15 hold row M=lane, K=k0+0/k0+1; lanes 16-31 hold K=k0+2/k0+3.
    // B (4x16 f32): lanes 0-15 hold col N=lane, K=k0+0/k0+1; lanes 16-31 hold K=k0+2/k0+3.
    {
        const int row16 = lane & 15;
        const int koff  = (lane >> 4) << 1;          // 0 or 2
#pragma unroll
        for (int tt = 0; tt < 2; ++tt) {
            int t  = wave * 2 + tt;                  // tile 0..15
            int ti = t >> 2, tj = t & 3;
            const float* aRow = &sm_mn[(ti * 16 + row16) * MN_STRIDE];
            const float* bRow = &sm_fn[(tj * 16 + row16) * MN_STRIDE];
            v8f acc = {};
#pragma unroll
            for (int k0 = 0; k0 < ND; k0 += 4) {
                v2f av = *(const v2f*)(aRow + k0 + koff);   // 8B aligned
                v2f bv = *(const v2f*)(bRow + k0 + koff);
                acc = __builtin_amdgcn_wmma_f32_16x16x4_f32(
                    false, av, false, bv, (short)0, acc, false, false);
            }
            // C/D layout: VGPR r -> M = r (+8 for lanes 16-31), N = lane&15
            int mbase = ti * 16 + ((lane >> 4) << 3);
            int n     = tj * 16 + (lane & 15);
#pragma unroll
            for (int r8 = 0; r8 < 8; ++r8)
                sm_sims[(mbase + r8) * SIMS_STRIDE + n] = acc[r8] * TEMP_INV;
        }
    }
    __syncthreads();

    // ---- Phase 4: per-row logsumexp - diag, accumulate over valid rows ----
    for (int r = wave; r < NF; r += 8) {
        float x0 = sm_sims[r * SIMS_STRIDE + lane];
        float x1 = sm_sims[r * SIMS_STRIDE + lane + 32];
        float mx = fmaxf(x0, x1);
#pragma unroll
        for (int m = 16; m >= 1; m >>= 1) mx = fmaxf(mx, __shfl_xor(mx, m, 32));
        float e = expf(x0 - mx) + expf(x1 - mx);
#pragma unroll
        for (int m = 16; m >= 1; m >>= 1) e += __shfl_xor(e, m, 32);
        if (lane == 0 && sm_cnt[r] > 0.0f) {
            float lse = mx + logf(e);
            float pos = sm_sims[r * SIMS_STRIDE + r];
            atomicAdd(&sm_acc[0], lse - pos);
            atomicAdd(&sm_acc[1], 1.0f);
        }
    }
    __syncthreads();

    // ---- Phase 5: block partial -> global workspace ----
    if (tid == 0) {
        atomicAdd(&ws[0], sm_acc[0]);
        atomicAdd(&ws[1], sm_acc[1]);
    }
}

__global__ void ncl_finalize(const float* __restrict__ ws, float* __restrict__ out) {
    float total = ws[0];
    float nval  = ws[1];
    out[0] = (nval > 0.0f) ? (total / fmaxf(nval, 1.0f)) : 0.0f;
}

extern "C" void kernel_launch(void* const* d_in, const int* in_sizes, int n_in,
                              void* d_out, int out_size, void* d_ws, size_t ws_size,
                              hipStream_t stream) {
    const float* atom_embed = (const float*)d_in[0];   // [B,A,D] f32
    const float* frag_embed = (const float*)d_in[1];   // [B,F,D] f32
    const int*   atom2frag  = (const int*)d_in[2];     // [B,A] int
    float* out = (float*)d_out;
    float* ws  = (float*)d_ws;

    int nb = in_sizes[0] / (NA * ND);                  // batches (2048)

    ncl_init_ws<<<1, 32, 0, stream>>>(ws);
    ncl_main<<<nb, 256, 0, stream>>>(atom_embed, frag_embed, atom2frag, ws);
    ncl_finalize<<<1, 1, 0, stream>>>(ws, out);
}